// MultiHeadAttentionQuantum_65481071406693
// MI455X (gfx1250) — compile-verified
//
#include <hip/hip_runtime.h>
#include <math.h>

typedef __attribute__((ext_vector_type(2))) float v2f;
typedef __attribute__((ext_vector_type(8))) float v8f;

#define B_     16
#define S_     2048
#define E_     8
#define H_     2
#define DK_    4
#define WAVES  8
#define BLOCK  (WAVES * 32)
#define PROW   17   // padded P-tile row stride (kills stride-16 bank conflicts)
#define QBATCH 8    // query tiles processed concurrently per wave

// One block per (b,h). Flash-attention with V_WMMA_F32_16X16X4_F32 (dk==4==K).
// Scores are bounded (|s| <= 2 since q=k=cos in [-1,1], dk=4, scale 0.5), so
// softmax needs no running max: p = exp2(s * 0.5*log2e) directly, and the
// row-sum denominator is produced by the PV WMMA via a ones-column in B.
// QBATCH query tiles share each K/V tile load (amortizes all B-side LDS traffic).
__global__ __launch_bounds__(BLOCK) void attn_kernel(
    const float* __restrict__ x, const float* __restrict__ theta,
    float* __restrict__ ao /* [B,S,E] attention output (workspace) */)
{
    __shared__ float qk[S_ * DK_];             // 32 KB: Q==K==V tile for this (b,h)
    __shared__ float pscr[WAVES][16 * PROW];   // per-wave P-tile bounce buffer

    const int bh  = blockIdx.x;
    const int b   = bh >> 1;
    const int h   = bh & 1;
    const int tid = threadIdx.x;

    const float4 th = ((const float4*)theta)[h];   // this head's 4 RX angles

    // ---- Phase 1: q = cos(x + theta) -> LDS [S][4] ----
    for (int s = tid; s < S_; s += BLOCK) {
        const float4 xv = *(const float4*)(x + ((size_t)b * S_ + s) * E_ + h * DK_);
        float4 qv;
        qv.x = __cosf(xv.x + th.x);
        qv.y = __cosf(xv.y + th.y);
        qv.z = __cosf(xv.z + th.z);
        qv.w = __cosf(xv.w + th.w);
        *(float4*)(qk + s * DK_) = qv;
    }
    __syncthreads();

    // ---- Phase 2: attention; each wave owns 16 query tiles, 8 at a time ----
    const int wave = tid >> 5;
    const int lane = tid & 31;
    const int half = lane >> 4;    // C/D layout: lanes 0-15 -> rows r, 16-31 -> r+8
    const int ln   = lane & 15;
    float* myp = &pscr[wave][0];

    const float sc = 0.5f * 1.44269504088896340736f;  // (1/sqrt(dk)) * log2(e)

    for (int qg = 0; qg < (S_ / 16) / (WAVES * QBATCH); ++qg) {
        // A operands (Q, 16x4 f32): M=ln; lanes 0-15 hold K={0,1}, 16-31 K={2,3}.
        v2f aq[QBATCH];
        v8f acc[QBATCH];   // D layout: cols 0..3 = P@V, col 4 = softmax denominator
        #pragma unroll
        for (int j = 0; j < QBATCH; ++j) {
            const int qbase = (wave + WAVES * (qg * QBATCH + j)) * 16;
            aq[j].x = qk[(qbase + ln) * DK_ + 2 * half + 0] * sc;
            aq[j].y = qk[(qbase + ln) * DK_ + 2 * half + 1] * sc;
            acc[j] = (v8f){};
        }

        for (int kt = 0; kt < S_ / 16; ++kt) {
            const int kbase = kt * 16;

            // B (K^T, 4x16 f32): N=ln; lanes 0-15 hold K={0,1}, 16-31 K={2,3}.
            v2f bk;
            bk.x = qk[(kbase + ln) * DK_ + 2 * half + 0];
            bk.y = qk[(kbase + ln) * DK_ + 2 * half + 1];

            // Hoisted [V | 1] B-operands for the 4 K=4 chunks (shared by QBATCH).
            v2f bv[4];
            #pragma unroll
            for (int c = 0; c < 4; ++c) {
                const int krow = kbase + c * 4 + 2 * half;
                const float v0 = qk[(krow + 0) * DK_ + (ln & 3)];
                const float v1 = qk[(krow + 1) * DK_ + (ln & 3)];
                bv[c].x = (ln < DK_) ? v0 : (ln == DK_ ? 1.0f : 0.0f);
                bv[c].y = (ln < DK_) ? v1 : (ln == DK_ ? 1.0f : 0.0f);
            }

            #pragma unroll
            for (int j = 0; j < QBATCH; ++j) {
                v8f cz = {};
                v8f st = __builtin_amdgcn_wmma_f32_16x16x4_f32(
                    false, aq[j], false, bk, (short)0, cz, false, false);

                // p = 2^(log2e * s); no max-shift needed (scores bounded).
                #pragma unroll
                for (int r = 0; r < 8; ++r)
                    st[r] = __builtin_amdgcn_exp2f(st[r]);

                // Bounce P (D layout) through wave-private LDS -> A layout.
                // Same-wave LDS ops are in-order: the reloads below see these
                // stores without an explicit drain.
                #pragma unroll
                for (int r = 0; r < 8; ++r)
                    myp[(r + 8 * half) * PROW + ln] = st[r];

                // Batch all 4 A-layout reloads, then run the 4 PV WMMAs so the
                // DS pipeline drains incrementally instead of wait-0 per chunk.
                v2f ap[4];
                #pragma unroll
                for (int c = 0; c < 4; ++c) {
                    ap[c].x = myp[ln * PROW + c * 4 + 2 * half + 0];
                    ap[c].y = myp[ln * PROW + c * 4 + 2 * half + 1];
                }
                #pragma unroll
                for (int c = 0; c < 4; ++c)
                    acc[j] = __builtin_amdgcn_wmma_f32_16x16x4_f32(
                        false, ap[c], false, bv[c], (short)0, acc[j], false, false);
            }
        }

        // Normalize by the ones-column (col 4) and write the 16x4 output slabs.
        #pragma unroll
        for (int j = 0; j < QBATCH; ++j) {
            const int qbase = (wave + WAVES * (qg * QBATCH + j)) * 16;
            #pragma unroll
            for (int r = 0; r < 8; ++r) {
                const float l    = __shfl(acc[j][r], half * 16 + DK_, 32);
                const float linv = __builtin_amdgcn_rcpf(l);
                if (ln < DK_) {
                    const int srow = qbase + r + 8 * half;
                    ao[((size_t)b * S_ + srow) * E_ + h * DK_ + ln] = acc[j][r] * linv;
                }
            }
        }
    }
}

// out[b,s,:] = ao[b,s,:] @ w_out^T + b_out   (E=8)
__global__ __launch_bounds__(256) void proj_kernel(
    const float* __restrict__ ao, const float* __restrict__ w,
    const float* __restrict__ bias, float* __restrict__ out)
{
    __shared__ float wl[E_ * E_];
    __shared__ float bl[E_];
    const int tid = threadIdx.x;
    if (tid < E_ * E_) wl[tid] = w[tid];
    if (tid < E_)      bl[tid] = bias[tid];
    __syncthreads();

    const size_t row = (size_t)blockIdx.x * 256 + tid;   // 0 .. B*S-1 (exact grid)
    const float4 a0 = *(const float4*)(ao + row * E_);
    const float4 a1 = *(const float4*)(ao + row * E_ + 4);
    const float av[8] = {a0.x, a0.y, a0.z, a0.w, a1.x, a1.y, a1.z, a1.w};

    float o[8];
    #pragma unroll
    for (int e = 0; e < E_; ++e) {
        float s = bl[e];
        #pragma unroll
        for (int k = 0; k < E_; ++k) s = fmaf(av[k], wl[e * E_ + k], s);
        o[e] = s;
    }
    *(float4*)(out + row * E_)     = make_float4(o[0], o[1], o[2], o[3]);
    *(float4*)(out + row * E_ + 4) = make_float4(o[4], o[5], o[6], o[7]);
}

extern "C" void kernel_launch(void* const* d_in, const int* in_sizes, int n_in,
                              void* d_out, int out_size, void* d_ws, size_t ws_size,
                              hipStream_t stream) {
    const float* x     = (const float*)d_in[0];   // [16,2048,8]
    const float* theta = (const float*)d_in[1];   // [8]
    const float* w_out = (const float*)d_in[2];   // [8,8]
    const float* b_out = (const float*)d_in[3];   // [8]
    float* out = (float*)d_out;                   // [16,2048,8]
    float* ao  = (float*)d_ws;                    // 1 MB attention output

    attn_kernel<<<B_ * H_, BLOCK, 0, stream>>>(x, theta, ao);
    proj_kernel<<<(B_ * S_) / 256, 256, 0, stream>>>(ao, w_out, b_out, out);
}